// GraphMultiComponentEncoder_54228257079798
// MI455X (gfx1250) — compile-verified
//
#include <hip/hip_runtime.h>
#include <hip/hip_bf16.h>

// Problem constants (match reference)
static constexpr int N_NODES = 100000;
static constexpr int N_EDGES = 800000;
static constexpr int H       = 128;     // F == H
static constexpr int G       = 128;
static constexpr int K       = 5;
static constexpr int OUT     = 64;
static constexpr int NL      = 5;

typedef float v2f __attribute__((ext_vector_type(2)));
typedef float v8f __attribute__((ext_vector_type(8)));

// ---------------------------------------------------------------------------
// 1) deg[n] = 1.0 (self loop)
__global__ __launch_bounds__(256) void init_deg_kernel(float* __restrict__ deg) {
    int i = blockIdx.x * 256 + threadIdx.x;
    if (i < N_NODES) deg[i] = 1.0f;
}

// 2) deg[dst] += valid(e)
__global__ __launch_bounds__(256) void edge_deg_kernel(const int* __restrict__ src,
                                                       const int* __restrict__ dst,
                                                       const int* __restrict__ batch,
                                                       const int* __restrict__ cluster,
                                                       float* __restrict__ deg) {
    int e = blockIdx.x * 256 + threadIdx.x;
    if (e >= N_EDGES) return;
    int s = src[e], d = dst[e];
    if (batch[s] == batch[d] && cluster[s] == cluster[d]) {
        unsafeAtomicAdd(&deg[d], 1.0f);
    }
}

// 3) dinv_sqrt[n] = rsqrt(deg[n])
__global__ __launch_bounds__(256) void node_rsqrt_kernel(const float* __restrict__ deg,
                                                         float* __restrict__ dinv) {
    int i = blockIdx.x * 256 + threadIdx.x;
    if (i < N_NODES) dinv[i] = rsqrtf(deg[i]);
}

// 4) norm[e] = dinv[src]*dinv[dst]*valid
__global__ __launch_bounds__(256) void edge_norm_kernel(const int* __restrict__ src,
                                                        const int* __restrict__ dst,
                                                        const int* __restrict__ batch,
                                                        const int* __restrict__ cluster,
                                                        const float* __restrict__ dinv,
                                                        float* __restrict__ norm) {
    int e = blockIdx.x * 256 + threadIdx.x;
    if (e >= N_EDGES) return;
    int s = src[e], d = dst[e];
    bool valid = (batch[s] == batch[d]) && (cluster[s] == cluster[d]);
    norm[e] = valid ? (dinv[s] * dinv[d]) : 0.0f;
}

// ---------------------------------------------------------------------------
// 5) GEMM t = h @ W  (N x 128 @ 128 x 128), fp32 WMMA 16x16x4.
//    Block: 256 threads = 8 waves; block owns 32 rows (two 16-row M-tiles);
//    wave w owns cols [16w,16w+16) and keeps two accumulators, reusing each
//    B fragment for two WMMAs. A tile staged into LDS with
//    GLOBAL_LOAD_ASYNC_TO_LDS_B128 (ASYNCcnt path).
//    Epilogue also writes agg = t * inv_deg (self-loop term): no agg memset.
__global__ __launch_bounds__(256) void gemm_self_kernel(const float* __restrict__ h,
                                                        const float* __restrict__ W,
                                                        const float* __restrict__ dinv,
                                                        float* __restrict__ t,
                                                        float* __restrict__ agg) {
    __shared__ float ldsA[32 * 132];          // pitch 132 -> conflict-free col reads
    const int tid     = threadIdx.x;
    const int lane    = tid & 31;
    const int wave    = tid >> 5;
    const int rowBase = blockIdx.x * 32;

    // Async staging of the 32x128 A tile: 1024 float4s, 4 per thread.
    {
        const unsigned ldsBase = (unsigned)(size_t)(&ldsA[0]);   // LDS offset = addr[31:0]
        for (int i = tid; i < 32 * 32; i += 256) {
            const int r  = i >> 5;
            const int c4 = i & 31;
            const unsigned ldsAddr = ldsBase + (unsigned)((r * 132 + c4 * 4) * 4);
            const float* gp = h + (size_t)(rowBase + r) * H + c4 * 4;
            asm volatile("global_load_async_to_lds_b128 %0, %1, off"
                         :: "v"(ldsAddr), "v"(gp) : "memory");
        }
        asm volatile("s_wait_asynccnt 0x0" ::: "memory");
    }
    __syncthreads();

    const int m     = lane & 15;   // row within M-tile (A), col within tile (B/D)
    const int khalf = lane >> 4;   // 0: K=k..k+1, 1: K=k+2..k+3  (ISA 16x4 f32 layout)
    const int col   = wave * 16 + m;

    v8f c0 = {0.f, 0.f, 0.f, 0.f, 0.f, 0.f, 0.f, 0.f};
    v8f c1 = {0.f, 0.f, 0.f, 0.f, 0.f, 0.f, 0.f, 0.f};
    for (int kk = 0; kk < H; kk += 4) {
        const int k0 = kk + khalf * 2;
        v2f a0, a1, b;
        a0[0] = ldsA[m * 132 + k0];
        a0[1] = ldsA[m * 132 + k0 + 1];
        a1[0] = ldsA[(m + 16) * 132 + k0];
        a1[1] = ldsA[(m + 16) * 132 + k0 + 1];
        b[0]  = W[(size_t)k0 * H + col];
        b[1]  = W[(size_t)(k0 + 1) * H + col];
        c0 = __builtin_amdgcn_wmma_f32_16x16x4_f32(false, a0, false, b,
                                                   (short)0, c0, false, false);
        c1 = __builtin_amdgcn_wmma_f32_16x16x4_f32(false, a1, false, b,
                                                   (short)0, c1, false, false);
    }

    // D layout: VGPR r -> lanes 0-15: M=r, lanes 16-31: M=r+8; N = lane&15.
#pragma unroll
    for (int r = 0; r < 8; ++r) {
        const int row0 = rowBase + r + khalf * 8;         // M-tile 0
        const int row1 = row0 + 16;                       // M-tile 1
        const float ds0 = dinv[row0];
        const float ds1 = dinv[row1];
        const size_t o0 = (size_t)row0 * H + col;
        const size_t o1 = (size_t)row1 * H + col;
        const float v0 = c0[r];
        const float v1 = c1[r];
        t[o0]   = v0;
        agg[o0] = v0 * (ds0 * ds0);   // inv_deg = dinv_sqrt^2
        t[o1]   = v1;
        agg[o1] = v1 * (ds1 * ds1);
    }
}

// ---------------------------------------------------------------------------
// 6) scatter: agg[dst,:] += t[src,:] * norm[e].  One wave per edge, float4/lane.
//    ~99.8% of edges have norm==0 -> uniform early exit.
__global__ __launch_bounds__(256) void scatter_kernel(const float* __restrict__ t,
                                                      const float* __restrict__ norm,
                                                      const int* __restrict__ src,
                                                      const int* __restrict__ dst,
                                                      float* __restrict__ agg) {
    const int lane = threadIdx.x & 31;
    const int e    = blockIdx.x * 8 + (threadIdx.x >> 5);
    const float nv = norm[e];
    if (nv == 0.0f) return;                 // wave-uniform branch
    const int s = src[e], d = dst[e];
    const float4 v = *(const float4*)(t + (size_t)s * H + lane * 4);
    float* base = agg + (size_t)d * H + lane * 4;
    unsafeAtomicAdd(base + 0, v.x * nv);
    unsafeAtomicAdd(base + 1, v.y * nv);
    unsafeAtomicAdd(base + 2, v.z * nv);
    unsafeAtomicAdd(base + 3, v.w * nv);
}

// 7) hOut = relu(agg + b)
__global__ __launch_bounds__(256) void bias_relu_kernel(const float* __restrict__ agg,
                                                        const float* __restrict__ b,
                                                        float* __restrict__ hOut) {
    const int i4 = blockIdx.x * 256 + threadIdx.x;      // float4 index
    const int f4 = i4 & 31;                             // 128/4 cols
    const float4 v  = *(const float4*)(agg + (size_t)i4 * 4);
    const float4 bb = *(const float4*)(b + f4 * 4);
    float4 r;
    r.x = fmaxf(v.x + bb.x, 0.0f);
    r.y = fmaxf(v.y + bb.y, 0.0f);
    r.z = fmaxf(v.z + bb.z, 0.0f);
    r.w = fmaxf(v.w + bb.w, 0.0f);
    *(float4*)(hOut + (size_t)i4 * 4) = r;
}

// ---------------------------------------------------------------------------
// 8) zero pooling buffers (sums: G*K*H, cnt: G*K)
__global__ __launch_bounds__(256) void zero_pool_kernel(float* __restrict__ sums,
                                                        float* __restrict__ cnt) {
    int i = blockIdx.x * 256 + threadIdx.x;
    if (i < G * K * H) sums[i] = 0.0f;
    if (i < G * K)     cnt[i]  = 0.0f;
}

// 9) pooled sums: one wave per node
__global__ __launch_bounds__(256) void pool_kernel(const float* __restrict__ h,
                                                   const int* __restrict__ batch,
                                                   const int* __restrict__ cluster,
                                                   float* __restrict__ sums,
                                                   float* __restrict__ cnt) {
    const int lane = threadIdx.x & 31;
    const int n    = blockIdx.x * 8 + (threadIdx.x >> 5);
    if (n >= N_NODES) return;
    const int seg = batch[n] * K + cluster[n];
    const float4 v = *(const float4*)(h + (size_t)n * H + lane * 4);
    float* base = sums + (size_t)seg * H + lane * 4;
    unsafeAtomicAdd(base + 0, v.x);
    unsafeAtomicAdd(base + 1, v.y);
    unsafeAtomicAdd(base + 2, v.z);
    unsafeAtomicAdd(base + 3, v.w);
    if (lane == 0) unsafeAtomicAdd(&cnt[seg], 1.0f);
}

// 10) mean in place: sums[i] /= max(cnt[i/H], 1)
__global__ __launch_bounds__(256) void mean_kernel(float* __restrict__ sums,
                                                   const float* __restrict__ cnt) {
    int i = blockIdx.x * 256 + threadIdx.x;
    if (i < G * K * H) sums[i] = sums[i] / fmaxf(cnt[i >> 7], 1.0f);
}

// 11) heads: mu = emb @ muW + mub ; logstd = emb @ lsW + lsb. emb: [G, K*H=640]
__global__ __launch_bounds__(64) void head_kernel(const float* __restrict__ emb,
                                                  const float* __restrict__ muW,
                                                  const float* __restrict__ mub,
                                                  const float* __restrict__ lsW,
                                                  const float* __restrict__ lsb,
                                                  float* __restrict__ out) {
    const int g = blockIdx.x;
    const int o = threadIdx.x;
    const float* e = emb + (size_t)g * (K * H);
    float am = 0.0f, al = 0.0f;
    for (int j = 0; j < K * H; ++j) {
        const float ev = e[j];
        am = fmaf(ev, muW[(size_t)j * OUT + o], am);
        al = fmaf(ev, lsW[(size_t)j * OUT + o], al);
    }
    out[(size_t)g * OUT + o]       = am + mub[o];
    out[(size_t)(G + g) * OUT + o] = al + lsb[o];   // logstd block after mu block
}

// ---------------------------------------------------------------------------
extern "C" void kernel_launch(void* const* d_in, const int* in_sizes, int n_in,
                              void* d_out, int out_size, void* d_ws, size_t ws_size,
                              hipStream_t stream) {
    const float* x       = (const float*)d_in[0];
    const float* convW   = (const float*)d_in[1];   // [NL, H, H]
    const float* convB   = (const float*)d_in[2];   // [NL, H]
    const float* muW     = (const float*)d_in[3];   // [K*H, OUT]
    const float* muB     = (const float*)d_in[4];
    const float* lsW     = (const float*)d_in[5];
    const float* lsB     = (const float*)d_in[6];
    const int*   eidx    = (const int*)d_in[7];     // [2, E]
    const int*   batch   = (const int*)d_in[8];
    const int*   cluster = (const int*)d_in[9];
    const int*   src = eidx;
    const int*   dst = eidx + N_EDGES;
    float* out = (float*)d_out;

    // workspace layout (floats)
    float* ws     = (float*)d_ws;
    float* deg    = ws;                              // N
    float* dinv   = deg  + N_NODES;                  // N
    float* norm   = dinv + N_NODES;                  // E
    float* tBuf   = norm + N_EDGES;                  // N*H
    float* aggBuf = tBuf   + (size_t)N_NODES * H;    // N*H
    float* hBuf   = aggBuf + (size_t)N_NODES * H;    // N*H
    float* sums   = hBuf   + (size_t)N_NODES * H;    // G*K*H
    float* cnt    = sums   + (size_t)G * K * H;      // G*K

    const int TB = 256;

    // degree / norm precompute
    init_deg_kernel<<<(N_NODES + TB - 1) / TB, TB, 0, stream>>>(deg);
    edge_deg_kernel<<<(N_EDGES + TB - 1) / TB, TB, 0, stream>>>(src, dst, batch, cluster, deg);
    node_rsqrt_kernel<<<(N_NODES + TB - 1) / TB, TB, 0, stream>>>(deg, dinv);
    edge_norm_kernel<<<(N_EDGES + TB - 1) / TB, TB, 0, stream>>>(src, dst, batch, cluster, dinv, norm);

    // 5 GCN layers
    const float* hIn = x;
    for (int l = 0; l < NL; ++l) {
        gemm_self_kernel<<<N_NODES / 32, TB, 0, stream>>>(
            hIn, convW + (size_t)l * H * H, dinv, tBuf, aggBuf);
        scatter_kernel<<<N_EDGES / 8, TB, 0, stream>>>(tBuf, norm, src, dst, aggBuf);
        bias_relu_kernel<<<(N_NODES * (H / 4)) / TB, TB, 0, stream>>>(
            aggBuf, convB + (size_t)l * H, hBuf);
        hIn = hBuf;
    }

    // pooling + heads
    zero_pool_kernel<<<(G * K * H + TB - 1) / TB, TB, 0, stream>>>(sums, cnt);
    pool_kernel<<<N_NODES / 8, TB, 0, stream>>>(hBuf, batch, cluster, sums, cnt);
    mean_kernel<<<(G * K * H + TB - 1) / TB, TB, 0, stream>>>(sums, cnt);
    head_kernel<<<G, OUT, 0, stream>>>(sums, muW, muB, lsW, lsB, out);
}